// Hard_Triplet_Loss_67731634258224
// MI455X (gfx1250) — compile-verified
//
#include <hip/hip_runtime.h>

#define N 512
#define F 32
#define B_TOT 128
#define KPAD 1024
#define BIG 1.0e9f
#define GAMMA 5.0f
#define INV_GAMMA 0.2f
#define MARGIN 1.0f

typedef float v2f __attribute__((ext_vector_type(2)));
typedef float v8f __attribute__((ext_vector_type(8)));

// ---------------------------------------------------------------- sq norms
__global__ __launch_bounds__(256) void sqnorm_kernel(const float* __restrict__ data,
                                                     float* __restrict__ sq) {
  int r = blockIdx.x * 256 + threadIdx.x;           // 65536 rows (b*512+i)
  const float4* p = (const float4*)(data + ((size_t)r << 5));
  float s = 0.f;
#pragma unroll
  for (int q = 0; q < 8; ++q) {
    float4 v = p[q];
    s += v.x * v.x + v.y * v.y + v.z * v.z + v.w * v.w;
  }
  sq[r] = s;
}

// ------------------------------------------------- WMMA gram -> diag-major D
// One wave per 16x16 tile (ti<=tj), mirror-stored. D laid out as Dd[b][k][i],
// k = i+j in [0,1023), padded stride 512 floats per diagonal.
__global__ __launch_bounds__(256) void gram_kernel(const float* __restrict__ data,
                                                   const float* __restrict__ sq,
                                                   float* __restrict__ Dd) {
  int wid  = blockIdx.x * 8 + (threadIdx.x >> 5);   // 128*528 waves total
  int lane = threadIdx.x & 31;
  int b = wid / 528;
  int t = wid - b * 528;
  // decode triangular tile index: (ti,tj), 0 <= ti <= tj < 32
  int ti = (int)((65.0f - sqrtf(4225.0f - 8.0f * (float)t)) * 0.5f);
  if (ti < 0) ti = 0;
  if (ti > 31) ti = 31;
  while (ti > 0 && (ti * 32 - (ti * (ti - 1)) / 2) > t) --ti;
  while (((ti + 1) * 32 - ((ti + 1) * ti) / 2) <= t) ++ti;
  int base = ti * 32 - (ti * (ti - 1)) / 2;
  int tj = ti + (t - base);

  int half = lane >> 4;                              // K half (0: k0,k1 / 1: k2,k3)
  int m    = lane & 15;
  const float* Xb   = data + ((size_t)b << 14);      // 512*32
  const float* Arow = Xb + (size_t)(ti * 16 + m) * F + half * 2;
  const float* Brow = Xb + (size_t)(tj * 16 + m) * F + half * 2;

  v8f c = {0.f, 0.f, 0.f, 0.f, 0.f, 0.f, 0.f, 0.f};
#pragma unroll
  for (int kk = 0; kk < F; kk += 4) {
    v2f a  = *(const v2f*)(Arow + kk);
    v2f bb = *(const v2f*)(Brow + kk);
    // D = A x B + C, fp32 16x16x4
    c = __builtin_amdgcn_wmma_f32_16x16x4_f32(false, a, false, bb, (short)0, c,
                                              false, false);
  }

  const float* sqb = sq + (b << 9);
  float* Db = Dd + ((size_t)b << 19);                // KPAD*N = 2^19 floats
  int gj = tj * 16 + m;
  float sqj = sqb[gj];
#pragma unroll
  for (int r = 0; r < 8; ++r) {
    int gi = ti * 16 + r + half * 8;                 // C layout: vgpr r, lane half
    float val = sqb[gi] + sqj - 2.0f * c[r];
    size_t k = (size_t)(gi + gj);
    Db[(k << 9) + gi] = val;                          // (i,j)
    Db[(k << 9) + gj] = val;                          // mirror (j,i)
  }
}

// ---------------------------------------------------------------- soft-DTW
__device__ __forceinline__ float lane_shift_up(float v, int lane) {
  int src = (lane - 1) & 31;
  return __int_as_float(__builtin_amdgcn_ds_bpermute(src << 2, __float_as_int(v)));
}

__device__ __forceinline__ float softmin3(float a, float b, float c) {
  float m = fminf(a, fminf(b, c));
  float s = __expf((m - a) * INV_GAMMA) + __expf((m - b) * INV_GAMMA) +
            __expf((m - c) * INV_GAMMA);
  return m - GAMMA * __logf(s);
}

// One step of the anti-diagonal recursion. RM2 (in) is overwritten with Rk.
#define DTW_STEP(RM2, RM1, KK)                                                  \
  {                                                                             \
    float rm2_im1 = lane_shift_up(RM2[15], lane);                               \
    float rm1_im1 = lane_shift_up(RM1[15], lane);                               \
    if (lane == 0) { rm2_im1 = BIG; rm1_im1 = BIG; }                            \
    const float* dbase = Db + ((size_t)(KK) << 9) + (lane << 4);                \
    float4 dq0 = *(const float4*)(dbase + 0);                                   \
    float4 dq1 = *(const float4*)(dbase + 4);                                   \
    float4 dq2 = *(const float4*)(dbase + 8);                                   \
    float4 dq3 = *(const float4*)(dbase + 12);                                  \
    float dv[16] = {dq0.x, dq0.y, dq0.z, dq0.w, dq1.x, dq1.y, dq1.z, dq1.w,     \
                    dq2.x, dq2.y, dq2.z, dq2.w, dq3.x, dq3.y, dq3.z, dq3.w};    \
    _Pragma("unroll")                                                           \
    for (int c = 0; c < 16; ++c) {                                              \
      int i = (lane << 4) + c;                                                  \
      int j = (KK) - i;                                                         \
      float rm1_i = RM1[c];                                                     \
      float rm2_i = RM2[c];                                                     \
      float rk = BIG;                                                           \
      if (j >= 0 && j < N) rk = dv[c] + softmin3(rm2_im1, rm1_im1, rm1_i);      \
      RM2[c] = rk;                                                              \
      rm2_im1 = rm2_i;                                                          \
      rm1_im1 = rm1_i;                                                          \
    }                                                                           \
  }

__global__ __launch_bounds__(32) void dtw_kernel(const float* __restrict__ Dd,
                                                 const int* __restrict__ lens,
                                                 float* __restrict__ dists) {
  int b = blockIdx.x;
  int lane = threadIdx.x;
  const float* Db = Dd + ((size_t)b << 19);
  int L = lens[b];
  L = L < 1 ? 1 : (L > N ? N : L);
  int kmax = 2 * L - 2;

  float X[16], Y[16];                                  // Rm2 / Rm1 ping-pong
#pragma unroll
  for (int c = 0; c < 16; ++c) { X[c] = BIG; Y[c] = BIG; }
  if (lane == 0) Y[0] = Db[0];                         // R0: only (0,0) valid

  for (int k = 1; k <= kmax;) {
    if (k + 2 <= kmax)
      __builtin_prefetch(Db + ((size_t)(k + 2) << 9) + (lane << 4), 0, 1);
    DTW_STEP(X, Y, k);                                 // odd k -> result in X
    ++k;
    if (k > kmax) break;
    DTW_STEP(Y, X, k);                                 // even k -> result in Y
    ++k;
  }

  int iL = L - 1;
  int tl = iL >> 4;
  int cL = iL & 15;
  if (lane == tl) {
    float v = BIG;
    if (kmax & 1) {
#pragma unroll
      for (int c = 0; c < 16; ++c)
        if (c == cL) v = X[c];
    } else {
#pragma unroll
      for (int c = 0; c < 16; ++c)
        if (c == cL) v = Y[c];
    }
    dists[b] = v / (2.0f * (float)L);
  }
}

// ----------------------------------------------------------- triplet loss
__global__ __launch_bounds__(32) void loss_kernel(const float* __restrict__ dists,
                                                  float* __restrict__ out) {
  __shared__ float acc[8];
  int w = threadIdx.x;
  if (w < 8) {
    float d[16];
#pragma unroll
    for (int s = 0; s < 16; ++s) d[s] = dists[w * 16 + s];
    float lv_sum = 0.f, nnz = 0.f;
#pragma unroll
    for (int i = 0; i < 6; ++i) {
      float mx = 0.f;
#pragma unroll
      for (int j = 0; j < 6; ++j) {
        float dpos = 0.5f * (d[i] + d[j]);
#pragma unroll
        for (int kk = 0; kk < 10; ++kk) {
          float dneg = 0.5f * (d[i] + d[6 + kk]);
          float sc = fmaxf(dpos + MARGIN - dneg, 0.0f);
          mx = fmaxf(mx, sc);
        }
      }
      lv_sum += mx;
      nnz += (mx != 0.0f) ? 1.0f : 0.0f;
    }
    lv_sum *= 60.0f;                                   // * g*NF
    nnz *= 60.0f;
    float lv = lv_sum / (nnz + 1.0f);
    float op = 0.0f;
#pragma unroll
    for (int i = 1; i < 6; ++i)
#pragma unroll
      for (int j = 0; j < i; ++j) op += 0.5f * (d[i] + d[j]);
    op *= 0.002f;                                      // MODEL_LAMBDA / NG
    acc[w] = lv + op;
  }
  __syncthreads();
  if (w == 0) {
    float s = 0.f;
    for (int q = 0; q < 8; ++q) s += acc[q];
    out[0] = s * 0.125f;                               // / NW
  }
}

// ---------------------------------------------------------------- launcher
extern "C" void kernel_launch(void* const* d_in, const int* in_sizes, int n_in,
                              void* d_out, int out_size, void* d_ws, size_t ws_size,
                              hipStream_t stream) {
  const float* data = (const float*)d_in[0];   // (128, 512, 32) fp32
  const int* lens   = (const int*)d_in[1];     // (128,) int32
  float* out = (float*)d_out;                  // scalar fp32

  float* sq    = (float*)d_ws;                 // 65536 floats
  float* dists = sq + 65536;                   // 128 floats
  float* Dd    = sq + 65792;                   // 128 * 1024 * 512 floats (diag-major)

  sqnorm_kernel<<<256, 256, 0, stream>>>(data, sq);
  gram_kernel<<<(B_TOT * 528) / 8, 256, 0, stream>>>(data, sq, Dd);
  dtw_kernel<<<B_TOT, 32, 0, stream>>>(Dd, lens, dists);
  loss_kernel<<<1, 32, 0, stream>>>(dists, out);
}